// KalmanFilterLayer_84464826843522
// MI455X (gfx1250) — compile-verified
//
#include <hip/hip_runtime.h>

#define NS 16
#define NM 8
#define NO 8

typedef __attribute__((ext_vector_type(2))) float v2f;
typedef __attribute__((ext_vector_type(8))) float v8f;

// One 16x16x16 f32 GEMM = 4 chained V_WMMA_F32_16X16X4_F32 (K = 4k + j + 2h).
__device__ __forceinline__ v8f wmma16(const v2f a[4], const v2f b[4], v8f c) {
#pragma unroll
  for (int k = 0; k < 4; ++k)
    c = __builtin_amdgcn_wmma_f32_16x16x4_f32(false, a[k], false, b[k],
                                              (short)0, c, false, false);
  return c;
}

// ---- LDS staging between chained GEMMs (D layout -> A/B layouts) ----
// D layout: VGPR v, lane l holds X[v + 8*(l>>4)][l&15]
__device__ __forceinline__ void stageD(float* sm, int n, int h, v8f d) {
#pragma unroll
  for (int v = 0; v < 8; ++v) sm[(v + 8 * h) * 17 + n] = d[v];
}
// B layout: chunk k, VGPR j, lane l -> X[4k + j + 2h][l&15]
__device__ __forceinline__ void loadB(const float* sm, int n, int h, v2f b[4]) {
#pragma unroll
  for (int k = 0; k < 4; ++k) {
    int k0 = 4 * k + 2 * h;
    b[k].x = sm[k0 * 17 + n];
    b[k].y = sm[(k0 + 1) * 17 + n];
  }
}
// A layout: chunk k, VGPR j, lane l -> X[l&15][4k + j + 2h]
__device__ __forceinline__ void loadA(const float* sm, int n, int h, v2f a[4]) {
#pragma unroll
  for (int k = 0; k < 4; ++k) {
    int k0 = 4 * k + 2 * h;
    a[k].x = sm[n * 17 + k0];
    a[k].y = sm[n * 17 + k0 + 1];
  }
}

// Kernel 1: per wave / per batch: P_pred = phi P0 phi^T + Q,
// GP = [H;C] P_pred, M = GP H^T (+R top-left). Writes S (8x8) and
// CPHt = C P_pred H^T (8x8) to workspace (128 floats per batch).
__global__ __launch_bounds__(256) void kf_wmma_kernel(
    const float* __restrict__ P0, const float* __restrict__ phi,
    const float* __restrict__ H, const float* __restrict__ Cm,
    const float* __restrict__ Qd, const float* __restrict__ Rd,
    float* __restrict__ ws, int nbatch) {
  __shared__ float smem_all[8][16 * 17];
  const int lane = threadIdx.x & 31;
  float* sm = &smem_all[threadIdx.x >> 5][0];
  const int n = lane & 15;  // column (B/D) or row (A)
  const int h = lane >> 4;  // lane half

  // Hoisted shared operands. Note: phi^T-as-B has the SAME register image as
  // phi-as-A, so aPhi serves both roles.
  v2f aPhi[4], aG[4], bHt[4];
#pragma unroll
  for (int k = 0; k < 4; ++k) {
    const int k0 = 4 * k + 2 * h;
    aPhi[k].x = phi[n * NS + k0];
    aPhi[k].y = phi[n * NS + k0 + 1];
    const float* grow = (n < 8) ? &H[n * NS] : &Cm[(n - 8) * NS];  // G=[H;C]
    aG[k].x = grow[k0];
    aG[k].y = grow[k0 + 1];
    // H^T as B: B[K][N] = H[N][K], columns N>=8 zero-padded
    bHt[k].x = (n < 8) ? H[n * NS + k0] : 0.f;
    bHt[k].y = (n < 8) ? H[n * NS + k0 + 1] : 0.f;
  }
  v8f dQ, dR;
#pragma unroll
  for (int v = 0; v < 8; ++v) {
    const int m = v + 8 * h;
    dQ[v] = (m == n) ? Qd[m] : 0.f;
    dR[v] = (m == n && m < 8) ? Rd[m] : 0.f;
  }
  const v8f dz = {0.f, 0.f, 0.f, 0.f, 0.f, 0.f, 0.f, 0.f};

  const int wave = (int)(blockIdx.x * blockDim.x + threadIdx.x) >> 5;
  const int nwaves = (int)(gridDim.x * blockDim.x) >> 5;

  for (int b = wave; b < nbatch; b += nwaves) {
    const float* p0 = P0 + (size_t)b * (NS * NS);
    // prefetch next tile this wave will touch (global_prefetch_b8)
    {
      int bn = b + nwaves;
      const float* pn = P0 + (size_t)(bn < nbatch ? bn : b) * (NS * NS);
      __builtin_prefetch(pn + n * NS, 0, 3);
    }
    // P0 in A layout (8B-aligned pair loads)
    v2f aP0[4];
#pragma unroll
    for (int k = 0; k < 4; ++k)
      aP0[k] = *(const v2f*)(p0 + n * NS + 4 * k + 2 * h);

    // U = P0 * phi^T
    v8f dU = wmma16(aP0, aPhi, dz);
    stageD(sm, n, h, dU);
    v2f bU[4];
    loadB(sm, n, h, bU);

    // P_pred = phi * U + diag(Q)
    v8f dP = wmma16(aPhi, bU, dQ);
    stageD(sm, n, h, dP);
    v2f bP[4];
    loadB(sm, n, h, bP);

    // GP = [H;C] * P_pred
    v8f dGP = wmma16(aG, bP, dz);
    stageD(sm, n, h, dGP);
    v2f aGP[4];
    loadA(sm, n, h, aGP);

    // M = GP * H^T + diag(R) in top-left; rows 0-7 = S, rows 8-15 = C P H^T
    v8f dM = wmma16(aGP, bHt, dR);

    if (n < 8) {
      float* o = ws + (size_t)b * 128 + h * 64;  // h=0 -> S, h=1 -> CPHt
#pragma unroll
      for (int v = 0; v < 8; ++v) o[v * 8 + n] = dM[v];
    }
  }
}

// Kernel 2: one thread per batch: x_pred = phi x0, y = z - H x_pred,
// Cholesky-solve S w = y, out = C x_pred + CPHt w.
__global__ __launch_bounds__(256) void kf_solve_kernel(
    const float* __restrict__ z, const float* __restrict__ x0,
    const float* __restrict__ phi, const float* __restrict__ H,
    const float* __restrict__ Cm, const float* __restrict__ ws,
    float* __restrict__ out, int nbatch) {
  const int b = blockIdx.x * blockDim.x + threadIdx.x;
  if (b >= nbatch) return;

  float x0r[NS];
#pragma unroll
  for (int k = 0; k < NS; ++k) x0r[k] = x0[(size_t)b * NS + k];

  float xp[NS];
#pragma unroll
  for (int m = 0; m < NS; ++m) {
    float s = 0.f;
#pragma unroll
    for (int k = 0; k < NS; ++k) s = fmaf(phi[m * NS + k], x0r[k], s);
    xp[m] = s;
  }

  float y[NM];
#pragma unroll
  for (int j = 0; j < NM; ++j) {
    float s = z[(size_t)b * NM + j];
#pragma unroll
    for (int k = 0; k < NS; ++k) s = fmaf(-H[j * NS + k], xp[k], s);
    y[j] = s;
  }

  const float* wsb = ws + (size_t)b * 128;
  float a[NM][NM];
#pragma unroll
  for (int i = 0; i < NM; ++i)
#pragma unroll
    for (int j = 0; j < NM; ++j) a[i][j] = wsb[i * 8 + j];

  // In-register Cholesky (S is SPD: H P H^T + R, R>0)
  float invd[NM];
#pragma unroll
  for (int i = 0; i < NM; ++i) {
#pragma unroll
    for (int j = 0; j <= i; ++j) {
      float s = a[i][j];
#pragma unroll
      for (int k = 0; k < j; ++k) s = fmaf(-a[i][k], a[j][k], s);
      if (j == i) {
        float d = sqrtf(fmaxf(s, 1e-30f));
        a[i][i] = d;
        invd[i] = 1.0f / d;
      } else {
        a[i][j] = s * invd[j];
      }
    }
  }
  // forward: L v = y
  float v[NM];
#pragma unroll
  for (int i = 0; i < NM; ++i) {
    float s = y[i];
#pragma unroll
    for (int k = 0; k < i; ++k) s = fmaf(-a[i][k], v[k], s);
    v[i] = s * invd[i];
  }
  // backward: L^T w = v
  float w[NM];
#pragma unroll
  for (int i = NM - 1; i >= 0; --i) {
    float s = v[i];
#pragma unroll
    for (int k = i + 1; k < NM; ++k) s = fmaf(-a[k][i], w[k], s);
    w[i] = s * invd[i];
  }

#pragma unroll
  for (int i = 0; i < NO; ++i) {
    float s = 0.f;
#pragma unroll
    for (int k = 0; k < NS; ++k) s = fmaf(Cm[i * NS + k], xp[k], s);
#pragma unroll
    for (int j = 0; j < NM; ++j) s = fmaf(wsb[64 + i * 8 + j], w[j], s);
    out[(size_t)b * NO + i] = s;
  }
}

extern "C" void kernel_launch(void* const* d_in, const int* in_sizes, int n_in,
                              void* d_out, int out_size, void* d_ws,
                              size_t ws_size, hipStream_t stream) {
  const float* z   = (const float*)d_in[0];
  const float* x0  = (const float*)d_in[1];
  const float* P0  = (const float*)d_in[2];
  const float* phi = (const float*)d_in[3];
  const float* H   = (const float*)d_in[4];
  const float* C   = (const float*)d_in[5];
  const float* Qd  = (const float*)d_in[6];
  const float* Rd  = (const float*)d_in[7];
  float* out = (float*)d_out;
  float* ws  = (float*)d_ws;  // needs nbatch*128 floats (128 MB @ BS=262144)

  const int nbatch = in_sizes[0] / NM;

  kf_wmma_kernel<<<2048, 256, 0, stream>>>(P0, phi, H, C, Qd, Rd, ws, nbatch);
  kf_solve_kernel<<<(nbatch + 255) / 256, 256, 0, stream>>>(z, x0, phi, H, C,
                                                            ws, out, nbatch);
}